// _multiCodebookQuantization_80539226734969
// MI455X (gfx1250) — compile-verified
//
#include <hip/hip_runtime.h>
#include <hip/hip_bf16.h>
#include <math.h>

// ---------------------------------------------------------------------------
// MultiCodebookQuantization forward for MI455X (gfx1250, wave32, WMMA).
// All matmuls use V_WMMA_F32_16X16X4_F32 (D=32 => 8 chained WMMAs / tile).
// Bandwidth-bound on the two 268MB f32 outputs; epilogues fused; hot GEMM
// stages A/B panels in LDS (bank-conflict-free, padded stride 36 DWORDs).
// ---------------------------------------------------------------------------

typedef __attribute__((ext_vector_type(2))) float v2f;
typedef __attribute__((ext_vector_type(8))) float v8f;

#define M_GROUPS 4
#define K_CODES  2048
#define D_DIM    32
#define N_BATCH  8
#define HW       1024                    // 32*32
#define S_TOTAL  (N_BATCH * HW)          // 8192 spatial rows per group
#define BOUND_F  0.0010000073f           // sqrt(1e-6 + (2^-18)^2)

#define TILE_S   128                     // s-rows per block (8 waves x 16)
#define TILE_K   64                      // k-cols per block
#define LSTRIDE  36                      // padded LDS row stride (floats):
                                         //  - 16B-aligned rows (36*4=144B)
                                         //  - stride-36 frag reads are LDS
                                         //    bank-conflict-free over 16 lanes

static __device__ __forceinline__ v8f wmma_f32_16x16x4(v2f a, v2f b, v8f c) {
  // 8 args: (neg_a, A, neg_b, B, c_mod, C, reuse_a, reuse_b)
  return __builtin_amdgcn_wmma_f32_16x16x4_f32(false, a, false, b, (short)0, c,
                                               false, false);
}

// ---------------------------------------------------------------------------
// Kernel 1: mapped[m,k,c] = sum_d codebook[m,k,d] * wC[m,c,d]
// One wave per 16x16 output tile. grid = (2, K/16=128, M=4), block=32.
// ---------------------------------------------------------------------------
__global__ void k_map_gemm(const float* __restrict__ codebook,
                           const float* __restrict__ wC,
                           float* __restrict__ mapped) {
  const int lane = threadIdx.x;
  const int half = lane >> 4;
  const int sub  = lane & 15;
  const int c0 = blockIdx.x * 16;
  const int r0 = blockIdx.y * 16;   // k rows
  const int m  = blockIdx.z;
  const float* A  = codebook + (size_t)(m * K_CODES + r0) * D_DIM;
  const float* Bt = wC       + (size_t)(m * D_DIM  + c0) * D_DIM;
  v8f acc = 0.0f;
#pragma unroll
  for (int kk = 0; kk < 8; ++kk) {
    const int dbase = kk * 4 + half * 2;
    v2f a, b;
    a[0] = A [sub * D_DIM + dbase]; a[1] = A [sub * D_DIM + dbase + 1];
    b[0] = Bt[sub * D_DIM + dbase]; b[1] = Bt[sub * D_DIM + dbase + 1];
    acc = wmma_f32_16x16x4(a, b, acc);
  }
  float* O = mapped + (size_t)(m * K_CODES) * D_DIM;
#pragma unroll
  for (int j = 0; j < 8; ++j) {
    const int r = r0 + j + half * 8;
    O[(size_t)r * D_DIM + (c0 + sub)] = acc[j];
  }
}

// ---------------------------------------------------------------------------
// Kernel 2: c2[m,k] = sum_d codebook[m,k,d]^2
// ---------------------------------------------------------------------------
__global__ void k_c2(const float* __restrict__ codebook, float* __restrict__ c2) {
  const int idx = blockIdx.x * blockDim.x + threadIdx.x;  // m*K + k
  const float* row = codebook + (size_t)idx * D_DIM;
  float s = 0.0f;
#pragma unroll
  for (int d = 0; d < D_DIM; ++d) s = fmaf(row[d], row[d], s);
  c2[idx] = s;
}

// ---------------------------------------------------------------------------
// Kernel 3: grouped 1x1 conv  x5[m,s,e] = sum_d x[n, m*32+d, p] * conv_w[m,e,d] + b
//   (s = n*1024 + p). One wave per 16x16 tile. grid = (2, S/16=512, 4).
// ---------------------------------------------------------------------------
__global__ void k_conv_gemm(const float* __restrict__ x,
                            const float* __restrict__ conv_w,
                            const float* __restrict__ conv_b,
                            float* __restrict__ x5g) {
  const int lane = threadIdx.x;
  const int half = lane >> 4;
  const int sub  = lane & 15;
  const int e0 = blockIdx.x * 16;
  const int s0 = blockIdx.y * 16;
  const int m  = blockIdx.z;
  const int s  = s0 + sub;                // A row this lane reads
  const int n  = s >> 10;
  const int p  = s & 1023;
  const float* Bt = conv_w + (size_t)(m * D_DIM + e0) * D_DIM;
  v8f acc = 0.0f;
#pragma unroll
  for (int kk = 0; kk < 8; ++kk) {
    const int dbase = kk * 4 + half * 2;
    v2f a, b;
    a[0] = x[(size_t)(n * 128 + m * 32 + dbase    ) * HW + p];
    a[1] = x[(size_t)(n * 128 + m * 32 + dbase + 1) * HW + p];
    b[0] = Bt[sub * D_DIM + dbase];
    b[1] = Bt[sub * D_DIM + dbase + 1];
    acc = wmma_f32_16x16x4(a, b, acc);
  }
  const int e = e0 + sub;
  const float bias = conv_b[m * D_DIM + e];
#pragma unroll
  for (int j = 0; j < 8; ++j) {
    const int r = s0 + j + half * 8;
    x5g[((size_t)(m * S_TOTAL + r)) * D_DIM + e] = acc[j] + bias;
  }
}

// ---------------------------------------------------------------------------
// Kernel 4: x2[m,s] = sum_e x5[m,s,e]^2
// ---------------------------------------------------------------------------
__global__ void k_x2(const float* __restrict__ x5g, float* __restrict__ x2) {
  const int idx = blockIdx.x * blockDim.x + threadIdx.x;  // m*S + s
  const float* row = x5g + (size_t)idx * D_DIM;
  float s = 0.0f;
#pragma unroll
  for (int d = 0; d < D_DIM; ++d) s = fmaf(row[d], row[d], s);
  x2[idx] = s;
}

// ---------------------------------------------------------------------------
// Kernel 5: main GEMM + fused distance/temperature epilogue.
//   inter[s,k] = x5[m,s,:] . mapped[m,k,:]
//   logit[n,m,p,k] = -sqrt(max(x2+c2-2*inter,0)) * max(temp,BOUND)^2
// Block = 256 threads (8 waves), tile = 128(s) x 64(k).
//  - A panel (128x32) + B panel (64x32) staged in LDS via coalesced float4
//    loads; padded stride 36 floats => conflict-free fragment reads.
//  - Each wave computes a 16x64 strip (4 k-tiles, A frags reused 4x).
// grid = (K/64=32, S/128=64, M=4).
// ---------------------------------------------------------------------------
__global__ __launch_bounds__(256) void
k_logit_gemm(const float* __restrict__ x5g,
             const float* __restrict__ mapped,
             const float* __restrict__ x2ws,
             const float* __restrict__ c2ws,
             const float* __restrict__ temperature,
             float* __restrict__ out_logit) {
  __shared__ float sA[TILE_S * LSTRIDE];   // 18 KB
  __shared__ float sB[TILE_K * LSTRIDE];   //  9 KB
  __shared__ float sX2[TILE_S];
  __shared__ float sC2[TILE_K];

  const int tid  = threadIdx.x;
  const int wave = tid >> 5;
  const int lane = tid & 31;
  const int half = lane >> 4;
  const int sub  = lane & 15;
  const int k0   = blockIdx.x * TILE_K;
  const int sblk = blockIdx.y * TILE_S;
  const int m    = blockIdx.z;

  const float4* Ag4 = (const float4*)(x5g    + (size_t)(m * S_TOTAL + sblk) * D_DIM);
  const float4* Bg4 = (const float4*)(mapped + (size_t)(m * K_CODES + k0)   * D_DIM);

  // Stage A panel: 128 rows x 8 float4 = 1024 float4; 4 per thread.
#pragma unroll
  for (int i = 0; i < 4; ++i) {
    const int idx = tid + 256 * i;
    const int row = idx >> 3, q = idx & 7;
    *(float4*)(sA + row * LSTRIDE + q * 4) = Ag4[idx];
  }
  // Stage B panel: 64 rows x 8 float4 = 512 float4; 2 per thread.
#pragma unroll
  for (int i = 0; i < 2; ++i) {
    const int idx = tid + 256 * i;
    const int row = idx >> 3, q = idx & 7;
    *(float4*)(sB + row * LSTRIDE + q * 4) = Bg4[idx];
  }
  if (tid < TILE_S) sX2[tid] = x2ws[m * S_TOTAL + sblk + tid];
  if (tid < TILE_K) sC2[tid] = c2ws[m * K_CODES + k0 + tid];
  __syncthreads();   // EXEC re-converged (all-ones) before WMMA region

  const float* Aw = sA + (wave * 16) * LSTRIDE;
  v8f acc0 = 0.0f, acc1 = 0.0f, acc2 = 0.0f, acc3 = 0.0f;
#pragma unroll
  for (int kk = 0; kk < 8; ++kk) {
    const int dbase = kk * 4 + half * 2;
    v2f a;
    a[0] = Aw[sub * LSTRIDE + dbase];
    a[1] = Aw[sub * LSTRIDE + dbase + 1];
    v2f b;
    const int bi = sub * LSTRIDE + dbase;
    b[0] = sB[bi];                  b[1] = sB[bi + 1];
    acc0 = wmma_f32_16x16x4(a, b, acc0);
    b[0] = sB[16 * LSTRIDE + bi];   b[1] = sB[16 * LSTRIDE + bi + 1];
    acc1 = wmma_f32_16x16x4(a, b, acc1);
    b[0] = sB[32 * LSTRIDE + bi];   b[1] = sB[32 * LSTRIDE + bi + 1];
    acc2 = wmma_f32_16x16x4(a, b, acc2);
    b[0] = sB[48 * LSTRIDE + bi];   b[1] = sB[48 * LSTRIDE + bi + 1];
    acc3 = wmma_f32_16x16x4(a, b, acc3);
  }

  float t = temperature[m];
  t = fmaxf(t, BOUND_F);
  const float t2 = t * t;

  v8f accs[4] = {acc0, acc1, acc2, acc3};
#pragma unroll
  for (int tt = 0; tt < 4; ++tt) {
    const int k = k0 + tt * 16 + sub;
    const float c2v = sC2[tt * 16 + sub];
#pragma unroll
    for (int j = 0; j < 8; ++j) {
      const int sl = wave * 16 + j + half * 8;   // row within block tile
      const int s  = sblk + sl;
      const int n  = s >> 10;
      const int p  = s & 1023;
      const float arg  = sX2[sl] + c2v - 2.0f * accs[tt][j];
      const float dist = sqrtf(fmaxf(arg, 0.0f));
      out_logit[((size_t)(n * M_GROUPS + m) * HW + p) * K_CODES + k] = -dist * t2;
    }
  }
}

// ---------------------------------------------------------------------------
// Kernel 6: Gumbel-max categorical sample per row (32768 rows of K=2048).
// One wave32 per row; shuffle tree reduction. NOTE: JAX threefry key(42)
// cannot be bit-reproduced; a counter-hash Gumbel draw stands in (same
// distribution, deterministic per (row,k)).
// ---------------------------------------------------------------------------
static __device__ __forceinline__ unsigned hash_u32(unsigned x) {
  x ^= x >> 16; x *= 0x7feb352du;
  x ^= x >> 15; x *= 0x846ca68bu;
  x ^= x >> 16;
  return x;
}

__global__ void k_sample(const float* __restrict__ logit, int* __restrict__ code) {
  const int wave = threadIdx.x >> 5;        // 8 waves/block, one row per wave
  const int lane = threadIdx.x & 31;
  const int row  = blockIdx.x * 8 + wave;   // flattened (n,m,h,w)
  const float* lrow = logit + (size_t)row * K_CODES;

  float best = -INFINITY;
  int   bidx = 0;
  for (int k = lane; k < K_CODES; k += 32) {
    if (k + 256 < K_CODES)
      __builtin_prefetch(lrow + k + 256, 0, 0);   // global_prefetch_b8
    const unsigned h =
        hash_u32((unsigned)row * 2654435761u ^ (unsigned)k * 2246822519u ^ 0x9E3779B9u);
    const float u = fmaf((float)(h >> 8), 1.0f / 16777216.0f, 1e-12f);
    const float g = -logf(-logf(u));
    const float v = lrow[k] + g;
    if (v > best || (v == best && k < bidx)) { best = v; bidx = k; }
  }
#pragma unroll
  for (int off = 16; off >= 1; off >>= 1) {
    const float ov = __shfl_xor(best, off, 32);
    const int   oi = __shfl_xor(bidx, off, 32);
    if (ov > best || (ov == best && oi < bidx)) { best = ov; bidx = oi; }
  }
  if (lane == 0) code[row] = bidx;
}

// ---------------------------------------------------------------------------
// Kernel 7: sample = one_hot(code)  (straight-through forward value is exactly
// the hard one-hot since hard + probs - probs == hard). float4 stores.
// ---------------------------------------------------------------------------
__global__ void k_onehot(const int* __restrict__ code, float4* __restrict__ sample) {
  const size_t i = (size_t)blockIdx.x * blockDim.x + threadIdx.x;  // 4 elems each
  const size_t elem = i * 4;
  const int row = (int)(elem >> 11);   // / K_CODES
  const int k   = (int)(elem & 2047);
  const int c   = code[row];
  float4 v;
  v.x = (k     == c) ? 1.0f : 0.0f;
  v.y = (k + 1 == c) ? 1.0f : 0.0f;
  v.z = (k + 2 == c) ? 1.0f : 0.0f;
  v.w = (k + 3 == c) ? 1.0f : 0.0f;
  sample[i] = v;
}

// ---------------------------------------------------------------------------
extern "C" void kernel_launch(void* const* d_in, const int* in_sizes, int n_in,
                              void* d_out, int out_size, void* d_ws, size_t ws_size,
                              hipStream_t stream) {
  (void)in_sizes; (void)n_in; (void)out_size; (void)ws_size;

  const float* x           = (const float*)d_in[0];
  const float* codebook    = (const float*)d_in[1];
  const float* wC          = (const float*)d_in[2];
  const float* conv_w      = (const float*)d_in[3];
  const float* conv_b      = (const float*)d_in[4];
  const float* temperature = (const float*)d_in[5];

  const size_t LOGIT_ELEMS = (size_t)N_BATCH * M_GROUPS * HW * K_CODES;  // 67108864
  const size_t CODE_ELEMS  = (size_t)N_BATCH * M_GROUPS * HW;            // 32768
  float* sample = (float*)d_out;
  int*   code   = (int*)((float*)d_out + LOGIT_ELEMS);
  float* logit  = (float*)d_out + LOGIT_ELEMS + CODE_ELEMS;

  char* ws = (char*)d_ws;
  float* x5g    = (float*)ws; ws += sizeof(float) * M_GROUPS * S_TOTAL * D_DIM;  // 4 MB
  float* mapped = (float*)ws; ws += sizeof(float) * M_GROUPS * K_CODES * D_DIM;  // 1 MB
  float* x2ws   = (float*)ws; ws += sizeof(float) * M_GROUPS * S_TOTAL;          // 128 KB
  float* c2ws   = (float*)ws; ws += sizeof(float) * M_GROUPS * K_CODES;          // 32 KB

  k_map_gemm  <<<dim3(2, K_CODES / 16, M_GROUPS), 32, 0, stream>>>(codebook, wC, mapped);
  k_c2        <<<(M_GROUPS * K_CODES) / 256, 256, 0, stream>>>(codebook, c2ws);
  k_conv_gemm <<<dim3(2, S_TOTAL / 16, M_GROUPS), 32, 0, stream>>>(x, conv_w, conv_b, x5g);
  k_x2        <<<(M_GROUPS * S_TOTAL) / 256, 256, 0, stream>>>(x5g, x2ws);
  k_logit_gemm<<<dim3(K_CODES / TILE_K, S_TOTAL / TILE_S, M_GROUPS), 256, 0, stream>>>(
      x5g, mapped, x2ws, c2ws, temperature, logit);
  k_sample    <<<(int)(CODE_ELEMS / 8), 256, 0, stream>>>(logit, code);
  k_onehot    <<<(int)(LOGIT_ELEMS / 4 / 256), 256, 0, stream>>>(code, (float4*)sample);
}